// GCN_83640193122405
// MI455X (gfx1250) — compile-verified
//
#include <hip/hip_runtime.h>

typedef __attribute__((ext_vector_type(2))) float v2f;
typedef __attribute__((ext_vector_type(8))) float v8f;

#define N_NODES 100000

// ---------------------------------------------------------------- utilities

__global__ __launch_bounds__(256) void zero_f4(float4* p, int n4) {
    int i = blockIdx.x * 256 + threadIdx.x;
    if (i < n4) p[i] = make_float4(0.f, 0.f, 0.f, 0.f);
}

// out[i][c] = b[c] for the 100000 x 32 output (bias pre-seeded before scatter)
__global__ __launch_bounds__(256) void bias_init32(float* out, const float* __restrict__ b) {
    int i = blockIdx.x * 256 + threadIdx.x;
    if (i < N_NODES * 32) out[i] = b[i & 31];
}

// ------------------------------------------------- edge scatter-add (segment sum)
// D floats per node (multiple of 4). D/4 lanes cooperate on one edge; each lane
// moves one float4 and issues 4 hardware global_atomic_add_f32 ops. Feature
// tables are L2-resident (25.6 MB << 192 MB L2), so gathers + atomics stay on-chip.

template<int D>
__global__ __launch_bounds__(256) void scatter_add(const float* __restrict__ X,
                                                   const int* __restrict__ src,
                                                   const int* __restrict__ dst,
                                                   float* out, int n_edges) {
    constexpr int LPE = D / 4;                    // lanes per edge (power of two)
    unsigned t = blockIdx.x * 256u + threadIdx.x;
    unsigned e = t / LPE;
    unsigned c = (t % LPE) * 4u;
    if (e >= (unsigned)n_edges) return;
    int s = src[e];
    int d = dst[e];
    float4 v = *(const float4*)(X + (size_t)s * D + c);
    float* o = out + (size_t)d * D + c;
    unsafeAtomicAdd(o + 0, v.x);
    unsafeAtomicAdd(o + 1, v.y);
    unsafeAtomicAdd(o + 2, v.z);
    unsafeAtomicAdd(o + 3, v.w);
}

// ------------------------------------------------- fp32 WMMA GEMM, K = 64
// out[N_NODES x NOUT] = (RELU?)(A[N_NODES x 64] @ W[64 x NOUT] (+ bias))
// One wave per 16x16 output tile; V_WMMA_F32_16X16X4_F32, 16 steps over K.

template<int NOUT, bool BIAS, bool RELU>
__global__ __launch_bounds__(256) void gemm_wmma(const float* __restrict__ A,
                                                 const float* __restrict__ W,
                                                 const float* __restrict__ bias,
                                                 float* __restrict__ out) {
    constexpr int CT    = NOUT / 16;                  // column tiles
    constexpr int TILES = (N_NODES / 16) * CT;
    int wave = threadIdx.x >> 5;
    int lane = threadIdx.x & 31;
    int tile = blockIdx.x * 8 + wave;
    if (tile >= TILES) return;                        // wave-uniform; EXEC stays all-1s
    int mBase = (tile / CT) * 16;
    int nBase = (tile % CT) * 16;
    int lh = lane >> 4;                               // lane half: K split (0->K0,1 ; 1->K2,3)
    int ll = lane & 15;

    const float* arow = A + (size_t)(mBase + ll) * 64 + 2 * lh;
    const float* wcol = W + nBase + ll;

    v8f acc = {};
#pragma unroll
    for (int k = 0; k < 64; k += 4) {
        // A 16x4 tile: lane<16 holds K=k,k+1 in v0,v1; lane>=16 holds K=k+2,k+3
        v2f a = *(const v2f*)(arow + k);
        // B 4x16 tile: v0 lanes0-15 = row k, v0 lanes16-31 = row k+2, v1 = k+1 / k+3
        v2f b;
        b.x = wcol[(size_t)(k + 2 * lh) * NOUT];
        b.y = wcol[(size_t)(k + 2 * lh + 1) * NOUT];
        acc = __builtin_amdgcn_wmma_f32_16x16x4_f32(false, a, false, b,
                                                    (short)0, acc, false, false);
    }

    float bv = BIAS ? bias[nBase + ll] : 0.f;
#pragma unroll
    for (int i = 0; i < 8; i++) {
        // C/D layout: VGPR i -> row mBase + i + 8*lh, col nBase + ll
        float v = acc[i] + bv;
        if (RELU) v = fmaxf(v, 0.f);
        out[(size_t)(mBase + i + 8 * lh) * NOUT + nBase + ll] = v;
    }
}

// ---------------------------------------------------------------- launcher

extern "C" void kernel_launch(void* const* d_in, const int* in_sizes, int n_in,
                              void* d_out, int out_size, void* d_ws, size_t ws_size,
                              hipStream_t stream) {
    const float* feat = (const float*)d_in[0];
    const int*   src  = (const int*)d_in[1];
    const int*   dst  = (const int*)d_in[2];
    const float* W1   = (const float*)d_in[3];
    const float* b1   = (const float*)d_in[4];
    const float* W2   = (const float*)d_in[5];
    const float* b2   = (const float*)d_in[6];
    const float* W3   = (const float*)d_in[7];
    const float* b3   = (const float*)d_in[8];
    float* out = (float*)d_out;

    float* bufA = (float*)d_ws;                      // agg buffer   [100000 x 64]
    float* bufB = bufA + (size_t)N_NODES * 64;       // hidden state [100000 x 64]
    float* bufC = bufB + (size_t)N_NODES * 64;       // h2 @ W3      [100000 x 32]

    const int n_edges = in_sizes[1];

    const int zeroBlocks   = (N_NODES * 64 / 4 + 255) / 256;
    const int scat64Blocks = (n_edges * 16 + 255) / 256;
    const int scat32Blocks = (n_edges * 8 + 255) / 256;
    const int gemm64Blocks = ((N_NODES / 16) * 4 + 7) / 8;
    const int gemm32Blocks = ((N_NODES / 16) * 2 + 7) / 8;

    // Layer 1: agg(features) -> relu(@W1 + b1)
    zero_f4<<<zeroBlocks, 256, 0, stream>>>((float4*)bufA, N_NODES * 64 / 4);
    scatter_add<64><<<scat64Blocks, 256, 0, stream>>>(feat, src, dst, bufA, n_edges);
    gemm_wmma<64, true, true><<<gemm64Blocks, 256, 0, stream>>>(bufA, W1, b1, bufB);

    // Layer 2: agg(h1) -> relu(@W2 + b2)
    zero_f4<<<zeroBlocks, 256, 0, stream>>>((float4*)bufA, N_NODES * 64 / 4);
    scatter_add<64><<<scat64Blocks, 256, 0, stream>>>(bufB, src, dst, bufA, n_edges);
    gemm_wmma<64, true, true><<<gemm64Blocks, 256, 0, stream>>>(bufA, W2, b2, bufB);

    // Layer 3 (linearity trick): agg(h2) @ W3 + b3 == agg(h2 @ W3) + b3
    // Project 64->32 first, halving the scatter traffic.
    gemm_wmma<32, false, false><<<gemm32Blocks, 256, 0, stream>>>(bufB, W3, nullptr, bufC);
    bias_init32<<<(N_NODES * 32 + 255) / 256, 256, 0, stream>>>(out, b3);
    scatter_add<32><<<scat32Blocks, 256, 0, stream>>>(bufC, src, dst, out, n_edges);
}